// RPN_9835475108288
// MI455X (gfx1250) — compile-verified
//
#include <hip/hip_runtime.h>
#include <math.h>

// ---------------------------------------------------------------------------
// RPN proposal generation for MI455X (gfx1250, wave32).
//
// Output layout (float32), total 9 * 59850 = 538650 elements:
//   [0          , 4*59850)  boxes   (deltas applied to anchors)
//   [4*59850    , 8*59850)  anchors
//   [8*59850    , 9*59850)  max_iou (max over 2000 proposals, per anchor)
// ---------------------------------------------------------------------------

#define NA_P3   45600
#define NA_P4   11400
#define NA_P5   2850
#define NA_TOT  59850
#define NPROP   2000
#define SCALE_CLAMP 3.3322045f   // ln(224/8)

// ---------------------------------------------------------------------------
// Kernel A: generate anchors and apply deltas.  One thread per anchor index.
// Reference ordering: per level, locations flattened as l = i*H + j with
// x = stride*(j+0.5), y = stride*(i+0.5); 3 aspect ratios (0.5, 1.0, 2.0)
// innermost.
// ---------------------------------------------------------------------------
__global__ void __launch_bounds__(256)
gen_boxes_anchors(const float* __restrict__ d3,
                  const float* __restrict__ d4,
                  const float* __restrict__ d5,
                  float*       __restrict__ out)
{
    int k = blockIdx.x * 256 + threadIdx.x;
    if (k >= NA_TOT) return;

    int   li;      // index within level (also the deltas row)
    float s;       // FPN stride
    int   H;       // SHAPES[level][0]
    const float* dl;
    if (k < NA_P3)              { li = k;                 s = 8.f;  H = 100; dl = d3; }
    else if (k < NA_P3 + NA_P4) { li = k - NA_P3;         s = 16.f; H = 50;  dl = d4; }
    else                        { li = k - NA_P3 - NA_P4; s = 32.f; H = 25;  dl = d5; }

    int loc = li / 3;
    int ar  = li - loc * 3;           // 0 -> 0.5, 1 -> 1.0, 2 -> 2.0
    int j   = loc % H;                // fast dim of meshgrid flatten
    int i   = loc / H;
    float x = s * ((float)j + 0.5f);
    float y = s * ((float)i + 0.5f);

    const float base = 4.f * s;       // STRIDE_SCALE * stride
    const float R2   = 1.41421356237f;
    const float IR2  = 0.70710678119f;
    float w = (ar == 0) ? base * R2 : ((ar == 1) ? base : base * IR2);
    float h = (ar == 0) ? base * IR2 : ((ar == 1) ? base : base * R2);

    float ax0 = x - 0.5f * w, ay0 = y - 0.5f * h;
    float ax1 = x + 0.5f * w, ay1 = y + 0.5f * h;

    // apply_deltas (anchor center is exactly (x, y), dims are (w, h))
    float4 dd = ((const float4*)dl)[li];
    float dw  = fminf(dd.z, SCALE_CLAMP);
    float dh  = fminf(dd.w, SCALE_CLAMP);
    float ncx = x + w * dd.x;
    float ncy = y + h * dd.y;
    float nw  = w * __expf(dw);
    float nh  = h * __expf(dh);

    ((float4*)out)[k]          = make_float4(ncx - 0.5f * nw, ncy - 0.5f * nh,
                                             ncx + 0.5f * nw, ncy + 0.5f * nh);
    ((float4*)out)[NA_TOT + k] = make_float4(ax0, ay0, ax1, ay1);
}

// ---------------------------------------------------------------------------
// gfx1250 async global->LDS staging (ASYNCcnt-tracked DMA-style copy).
// LDS byte address = low 32 bits of the flat pointer to __shared__ memory.
// ---------------------------------------------------------------------------
__device__ __forceinline__ void async_copy_b128(void* lds_dst, const void* gsrc)
{
    unsigned int       lds = (unsigned int)(unsigned long long)(uintptr_t)lds_dst;
    unsigned long long g   = (unsigned long long)(uintptr_t)gsrc;
    asm volatile("global_load_async_to_lds_b128 %0, %1, off"
                 :: "v"(lds), "v"(g) : "memory");
}

__device__ __forceinline__ void wait_asynccnt0()
{
    asm volatile("s_wait_asynccnt 0" ::: "memory");
}

__device__ __forceinline__ float iou_term(float4 p,
                                          float ax0, float ay0, float ax1, float ay1,
                                          float aarea)
{
    float xl = fmaxf(ax0, p.x);
    float yl = fmaxf(ay0, p.y);
    float xr = fminf(ax1, p.z);
    float yr = fminf(ay1, p.w);
    float iw = fmaxf(xr - xl, 0.f);
    float ih = fmaxf(yr - yl, 0.f);
    float inter = iw * ih;
    float parea = (p.z - p.x) * (p.w - p.y);
    float denom = parea + aarea - inter;
    return inter * __builtin_amdgcn_rcpf(denom);   // v_rcp_f32, ~1 ulp
}

// ---------------------------------------------------------------------------
// Kernel B: max_iou[j] = max_i IoU(proposal_i, anchor_j).
// Proposals = boxes[0:2000] (32 KB) async-staged once per block into LDS
// (320 KB/WGP on CDNA5), then reused by all 256 anchors of the block via
// broadcast ds_load_b128 reads.  4 accumulators break the v_max chain.
// ---------------------------------------------------------------------------
__global__ void __launch_bounds__(256)
max_iou_kernel(const float* __restrict__ base,   // d_out base: boxes then anchors
               float*       __restrict__ max_out)
{
    __shared__ float4 sP[NPROP];                  // 32000 bytes

    const int tid = threadIdx.x;

    // Async-stage all 2000 proposal boxes (boxes region starts at base).
    for (int i = tid; i < NPROP; i += 256)
        async_copy_b128(&sP[i], base + 4 * i);
    wait_asynccnt0();
    __syncthreads();

    const int j = blockIdx.x * 256 + tid;
    if (j >= NA_TOT) return;

    float4 a = ((const float4*)base)[NA_TOT + j]; // anchors region
    float ax0 = a.x, ay0 = a.y, ax1 = a.z, ay1 = a.w;
    float aarea = (ax1 - ax0) * (ay1 - ay0);

    float m0 = 0.f, m1 = 0.f, m2 = 0.f, m3 = 0.f;
    #pragma unroll 2
    for (int i = 0; i < NPROP; i += 4) {          // NPROP % 4 == 0
        m0 = fmaxf(m0, iou_term(sP[i + 0], ax0, ay0, ax1, ay1, aarea));
        m1 = fmaxf(m1, iou_term(sP[i + 1], ax0, ay0, ax1, ay1, aarea));
        m2 = fmaxf(m2, iou_term(sP[i + 2], ax0, ay0, ax1, ay1, aarea));
        m3 = fmaxf(m3, iou_term(sP[i + 3], ax0, ay0, ax1, ay1, aarea));
    }
    max_out[j] = fmaxf(fmaxf(m0, m1), fmaxf(m2, m3));
}

// ---------------------------------------------------------------------------
extern "C" void kernel_launch(void* const* d_in, const int* in_sizes, int n_in,
                              void* d_out, int out_size, void* d_ws, size_t ws_size,
                              hipStream_t stream)
{
    (void)in_sizes; (void)n_in; (void)out_size; (void)d_ws; (void)ws_size;

    // setup_inputs order: feat_p3, feat_p4, feat_p5, deltas_p3, deltas_p4, deltas_p5
    const float* d3 = (const float*)d_in[3];
    const float* d4 = (const float*)d_in[4];
    const float* d5 = (const float*)d_in[5];
    float* out = (float*)d_out;

    dim3 blk(256);
    dim3 grd((NA_TOT + 255) / 256);   // 234 blocks

    gen_boxes_anchors<<<grd, blk, 0, stream>>>(d3, d4, d5, out);
    max_iou_kernel  <<<grd, blk, 0, stream>>>(out, out + 8 * (size_t)NA_TOT);
}